// TokenAdaptionModule_4260607557731
// MI455X (gfx1250) — compile-verified
//
#include <hip/hip_runtime.h>
#include <hip/hip_bf16.h>
#include <math.h>

// Problem constants (from reference)
#define BB    64
#define LL    4096
#define CC    512
#define KKP   2048      // kept tokens per batch
#define PTOK  819       // P
#define HH    102       // H
#define KP1   2049      // K+1 token rows
#define KPAD  2064      // token rows padded to multiple of 16
#define HP    112       // H padded to multiple of 16
#define PPAD  832       // P padded to multiple of 16
#define LNEPS 1e-5f

typedef float v2f __attribute__((ext_vector_type(2)));
typedef float v8f __attribute__((ext_vector_type(8)));

// ------------------------------------------------ pack & pad (k-pair layouts)
// w1pk : [(c>>2)*HP   + n]*4 + (c&3)   (so a lane's k-pair is one aligned b64)
// w2pk : [(j>>2)*PPAD + p]*4 + (j&3)
// lnp  : [(c>>1)*4 + {g0,g1,b0,b1}]    (one b128 gives LN coefs for a c-pair)
__global__ __launch_bounds__(256)
void tam_pack_kernel(const float* __restrict__ w1, const float* __restrict__ b1,
                     const float* __restrict__ w2, const float* __restrict__ b2,
                     const float* __restrict__ ln_g, const float* __restrict__ ln_b,
                     float* __restrict__ w1pk, float* __restrict__ lnp,
                     float* __restrict__ b1p, float* __restrict__ w2pk,
                     float* __restrict__ b2p, float* __restrict__ hpad) {
  int i = blockIdx.x * 256 + threadIdx.x;
  const int n1 = CC * HP;           // 57344  w1pk
  const int n2 = n1 + CC * 2;       // +1024  lnp
  const int n3 = n2 + HP;           // +112   b1p
  const int n4 = n3 + HP * PPAD;    // +93184 w2pk
  const int n5 = n4 + PPAD;         // +832   b2p
  const int n6 = n5 + 16 * HP;      // +1792  hbuf pad rows (zero)
  if (i < n1) {
    int c = i / HP, n = i % HP;
    w1pk[((size_t)(c >> 2) * HP + n) * 4 + (c & 3)] = (n < HH) ? w1[c * HH + n] : 0.f;
  } else if (i < n2) {
    int t = i - n1, pair = t >> 2, sub = t & 3;
    lnp[t] = (sub < 2) ? ln_g[pair * 2 + sub] : ln_b[pair * 2 + sub - 2];
  } else if (i < n3) {
    int j = i - n2;
    b1p[j] = (j < HH) ? b1[j] : 0.f;
  } else if (i < n4) {
    int t = i - n3, j = t / PPAD, p = t % PPAD;
    w2pk[((size_t)(j >> 2) * PPAD + p) * 4 + (j & 3)] =
        (j < HH && p < PTOK) ? w2[j * PTOK + p] : 0.f;
  } else if (i < n5) {
    int p = i - n4;
    b2p[p] = (p < PTOK) ? b2[p] : 0.f;
  } else if (i < n6) {
    hpad[i - n5] = 0.f;
  }
}

// ------------------------------------- exact rank (stable top-K) + softmax w
__global__ __launch_bounds__(1024)
void tam_rank_kernel(const float* __restrict__ ax, const float* __restrict__ ay,
                     int* __restrict__ rankbuf, float* __restrict__ wbuf,
                     float* __restrict__ mask_out) {
  __shared__ float s[LL];
  __shared__ unsigned char kf[LL];
  __shared__ float red[1024];
  const int b = blockIdx.x;
  const int t = threadIdx.x;
  for (int l = t; l < LL; l += 1024) s[l] = ax[b * LL + l] + ay[b * LL + l];
  __syncthreads();
  // rank = #elements strictly greater (ties broken by lower index) -> unique
  const float4* s4 = (const float4*)s;
  for (int l = t; l < LL; l += 1024) {
    float si = s[l];
    int cnt = 0;
    for (int j4 = 0; j4 < LL / 4; ++j4) {
      float4 v = s4[j4];
      int j = j4 * 4;
      cnt += (v.x > si) || (v.x == si && (j + 0) < l);
      cnt += (v.y > si) || (v.y == si && (j + 1) < l);
      cnt += (v.z > si) || (v.z == si && (j + 2) < l);
      cnt += (v.w > si) || (v.w == si && (j + 3) < l);
    }
    rankbuf[b * LL + l] = cnt;
    kf[l] = (cnt < KKP) ? 1 : 0;
    mask_out[b * LL + l] = (cnt < KKP) ? 1.0f : 0.0f;
  }
  __syncthreads();
  // softmax over non-kept scores
  float m = -INFINITY;
  for (int l = t; l < LL; l += 1024) if (!kf[l]) m = fmaxf(m, s[l]);
  red[t] = m; __syncthreads();
  for (int st = 512; st > 0; st >>= 1) {
    if (t < st) red[t] = fmaxf(red[t], red[t + st]);
    __syncthreads();
  }
  m = red[0]; __syncthreads();
  float z = 0.f;
  for (int l = t; l < LL; l += 1024) if (!kf[l]) z += expf(s[l] - m);
  red[t] = z; __syncthreads();
  for (int st = 512; st > 0; st >>= 1) {
    if (t < st) red[t] += red[t + st];
    __syncthreads();
  }
  float inv = 1.0f / red[0];
  for (int l = t; l < LL; l += 1024)
    wbuf[b * LL + l] = kf[l] ? 0.f : expf(s[l] - m) * inv;
}

// -------------------------- gather kept rows to slot=rank, fused LN statistics
__global__ __launch_bounds__(128)
void tam_gather_kernel(const float* __restrict__ tokens, const int* __restrict__ rankbuf,
                       float* __restrict__ all_t, float* __restrict__ mu,
                       float* __restrict__ rs) {
  const int blk = blockIdx.x;
  const int b = blk >> 12;       // / 4096
  const int l = blk & 4095;
  const int r = rankbuf[b * LL + l];
  if (r >= KKP) return;
  const float4* src = (const float4*)(tokens + ((size_t)b * LL + l) * CC);
  float4* dst = (float4*)(all_t + ((size_t)b * KPAD + r) * CC);
  const int t = threadIdx.x;
  float4 v = src[t];
  dst[t] = v;
  float sum = v.x + v.y + v.z + v.w;
  float sq  = v.x * v.x + v.y * v.y + v.z * v.z + v.w * v.w;
  __shared__ float r1[128], r2[128];
  r1[t] = sum; r2[t] = sq; __syncthreads();
  for (int st = 64; st > 0; st >>= 1) {
    if (t < st) { r1[t] += r1[t + st]; r2[t] += r2[t + st]; }
    __syncthreads();
  }
  if (t == 0) {
    float mean = r1[0] * (1.0f / CC);
    float var  = r2[0] * (1.0f / CC) - mean * mean;
    mu[b * KP1 + r] = mean;
    rs[b * KP1 + r] = rsqrtf(var + LNEPS);
  }
}

// ------------------------------------ extra token = softmax-weighted reduction
__global__ __launch_bounds__(256)
void tam_extra_kernel(const float* __restrict__ tokens, const float* __restrict__ wbuf,
                      float* __restrict__ all_t) {
  const int b = blockIdx.x;
  const int tx = threadIdx.x & 63;
  const int ty = threadIdx.x >> 6;           // 0..3 (uniform per wave pair)
  const int c = blockIdx.y * 64 + tx;
  float acc = 0.f;
  for (int l = ty; l < LL; l += 4) {
    float w = wbuf[b * LL + l];
    if (w != 0.f) acc += w * tokens[((size_t)b * LL + l) * CC + c];
  }
  __shared__ float red[256];
  red[threadIdx.x] = acc; __syncthreads();
  if (ty == 0) {
    acc = red[tx] + red[tx + 64] + red[tx + 128] + red[tx + 192];
    all_t[((size_t)b * KPAD + KKP) * CC + c] = acc;
  }
}

// ----------------------- LN stats for extra row + zero the padded token rows
__global__ __launch_bounds__(128)
void tam_extra_stats_kernel(float* __restrict__ all_t, float* __restrict__ mu,
                            float* __restrict__ rs) {
  const int b = blockIdx.x;
  const int t = threadIdx.x;
  const float4* row = (const float4*)(all_t + ((size_t)b * KPAD + KKP) * CC);
  float4 v = row[t];
  float sum = v.x + v.y + v.z + v.w;
  float sq  = v.x * v.x + v.y * v.y + v.z * v.z + v.w * v.w;
  __shared__ float r1[128], r2[128];
  r1[t] = sum; r2[t] = sq; __syncthreads();
  for (int st = 64; st > 0; st >>= 1) {
    if (t < st) { r1[t] += r1[t + st]; r2[t] += r2[t + st]; }
    __syncthreads();
  }
  if (t == 0) {
    float mean = r1[0] * (1.0f / CC);
    float var  = r2[0] * (1.0f / CC) - mean * mean;
    mu[b * KP1 + KKP] = mean;
    rs[b * KP1 + KKP] = rsqrtf(var + LNEPS);
  }
  float4 zero = make_float4(0.f, 0.f, 0.f, 0.f);
  float4* pad = (float4*)(all_t + ((size_t)b * KPAD + KP1) * CC);
  const int n4 = (KPAD - KP1) * CC / 4;       // 1920
  for (int i = t; i < n4; i += 128) pad[i] = zero;
}

// --------------- GEMM1: h = gelu(LN(all_tokens) @ w1 + b1)  [WMMA f32, 7 acc]
// One wave owns one 16-row M-tile x all 112 H columns: 7 wmma per A fragment.
// No row masking needed: invalid A rows only pollute invalid D rows (unstored).
__global__ __launch_bounds__(256)
void tam_gemm1_kernel(const float* __restrict__ all_t, const float* __restrict__ mu,
                      const float* __restrict__ rs, const float* __restrict__ lnp,
                      const float* __restrict__ w1pk, const float* __restrict__ b1p,
                      float* __restrict__ hbuf) {
  const int b = blockIdx.y;
  const int wave = threadIdx.x >> 5;
  const int lane = threadIdx.x & 31;
  const int mt = blockIdx.x * 8 + wave;
  if (mt >= KPAD / 16) return;                 // 129 M-tiles
  const int n  = lane & 15;
  const int kp = (lane >> 4) << 1;             // 0 or 2
  const int row = mt * 16 + n;
  const int rowc = (row < KP1) ? row : (KP1 - 1);
  const float* arow = all_t + ((size_t)b * KPAD + row) * CC;   // rows<2064 valid
  const float mym = mu[b * KP1 + rowc];
  const float myr = rs[b * KP1 + rowc];
  v8f acc[7];
#pragma unroll
  for (int t = 0; t < 7; ++t) for (int i = 0; i < 8; ++i) acc[t][i] = 0.f;
  for (int k0 = 0; k0 < CC; k0 += 4) {
    const int c0 = k0 + kp;
    const float4 ln = *(const float4*)(lnp + (size_t)(c0 >> 1) * 4);   // g0 g1 b0 b1
    const v2f av = *(const v2f*)(arow + c0);                            // b64
    v2f a;
    a.x = (av.x - mym) * myr * ln.x + ln.z;
    a.y = (av.y - mym) * myr * ln.y + ln.w;
    const float* bbase = w1pk + (size_t)(k0 >> 2) * HP * 4 + kp;
#pragma unroll
    for (int t = 0; t < 7; ++t) {
      const v2f bb = *(const v2f*)(bbase + (t * 16 + n) * 4);           // b64
      acc[t] = __builtin_amdgcn_wmma_f32_16x16x4_f32(false, a, false, bb,
                                                     (short)0, acc[t], false, false);
    }
  }
  const int mbase = (lane < 16) ? 0 : 8;
#pragma unroll
  for (int t = 0; t < 7; ++t) {
    const int ncol = t * 16 + n;
    const float bias = b1p[ncol];
    for (int r2 = 0; r2 < 8; ++r2) {
      const int mrow = mt * 16 + mbase + r2;
      if (mrow < KP1) {
        float x = acc[t][r2] + bias;
        hbuf[((size_t)b * KP1 + mrow) * HP + ncol] =
            0.5f * x * (1.0f + erff(x * 0.70710678118654752f));  // exact gelu
      }
    }
  }
}

// -------- GEMM2: logits = (h @ w2 + b2)*scale  [WMMA f32, 4-wide N blocking]
__global__ __launch_bounds__(256)
void tam_gemm2_kernel(const float* __restrict__ hbuf, const float* __restrict__ w2pk,
                      const float* __restrict__ b2p, const float* __restrict__ scale,
                      float* __restrict__ wgt) {
  const int b = blockIdx.y;
  const int wave = threadIdx.x >> 5;
  const int lane = threadIdx.x & 31;
  const int tile = blockIdx.x * 8 + wave;
  const int MT = KPAD / 16;                    // 129
  const int NG = (PPAD / 16) / 4;              // 13 groups of 4 N-tiles
  if (tile >= MT * NG) return;
  const int mt = tile / NG, g = tile % NG;
  const int n  = lane & 15;
  const int kp = (lane >> 4) << 1;
  const int row = mt * 16 + n;
  const float* arow = hbuf + ((size_t)b * KP1 + row) * HP;     // padded tail
  const float scl = scale[0];
  v8f acc[4];
#pragma unroll
  for (int t = 0; t < 4; ++t) for (int i = 0; i < 8; ++i) acc[t][i] = 0.f;
  for (int j0 = 0; j0 < HP; j0 += 4) {
    const v2f a = *(const v2f*)(arow + j0 + kp);                        // b64
    const float* bbase = w2pk + (size_t)(j0 >> 2) * PPAD * 4 + (g * 64 + n) * 4 + kp;
#pragma unroll
    for (int t = 0; t < 4; ++t) {
      const v2f bb = *(const v2f*)(bbase + t * 64);                     // b64
      acc[t] = __builtin_amdgcn_wmma_f32_16x16x4_f32(false, a, false, bb,
                                                     (short)0, acc[t], false, false);
    }
  }
  const int mbase = (lane < 16) ? 0 : 8;
#pragma unroll
  for (int t = 0; t < 4; ++t) {
    const int pcol = g * 64 + t * 16 + n;
    const float bias = b2p[pcol];
    for (int r2 = 0; r2 < 8; ++r2) {
      const int mrow = mt * 16 + mbase + r2;
      if (mrow < KP1)
        wgt[((size_t)b * KPAD + mrow) * PPAD + pcol] = (acc[t][r2] + bias) * scl;
    }
  }
}

// --------------------------- softmax over token axis (per (b,p)); zeros pads
__global__ __launch_bounds__(256)
void tam_softmax_kernel(float* __restrict__ wgt) {
  const int b = blockIdx.y;
  const int tx = threadIdx.x & 63;
  const int ty = threadIdx.x >> 6;             // 0..3
  const int p = blockIdx.x * 64 + tx;          // < 832
  __shared__ float red[256];
  float m = -INFINITY;
  for (int k = ty; k < KP1; k += 4)
    m = fmaxf(m, wgt[((size_t)b * KPAD + k) * PPAD + p]);
  red[threadIdx.x] = m; __syncthreads();
  m = fmaxf(fmaxf(red[tx], red[tx + 64]), fmaxf(red[tx + 128], red[tx + 192]));
  __syncthreads();
  float z = 0.f;
  for (int k = ty; k < KP1; k += 4)
    z += expf(wgt[((size_t)b * KPAD + k) * PPAD + p] - m);
  red[threadIdx.x] = z; __syncthreads();
  z = red[tx] + red[tx + 64] + red[tx + 128] + red[tx + 192];
  const float inv = 1.0f / z;
  for (int k = ty; k < KPAD; k += 4) {
    const size_t idx = ((size_t)b * KPAD + k) * PPAD + p;
    wgt[idx] = (k < KP1) ? expf(wgt[idx] - m) * inv : 0.f;
  }
}

// ------ final einsum: super = wgt^T(p,k) @ all_tokens  [WMMA f32, 8-wide N]
__global__ __launch_bounds__(256)
void tam_einsum_kernel(const float* __restrict__ wgt, const float* __restrict__ all_t,
                       float* __restrict__ super) {
  const int b = blockIdx.y;
  const int wave = threadIdx.x >> 5;
  const int lane = threadIdx.x & 31;
  const int tile = blockIdx.x * 8 + wave;
  const int MT = PPAD / 16;                    // 52
  const int NG = (CC / 16) / 8;                // 4 groups of 8 C-tiles
  if (tile >= MT * NG) return;
  const int mt = tile >> 2, g = tile & 3;
  const int n  = lane & 15;
  const int kp = (lane >> 4) << 1;
  const int pidx = mt * 16 + n;                // A row = p
  const int cbase = g * 128;
  v8f acc[8];
#pragma unroll
  for (int t = 0; t < 8; ++t) for (int i = 0; i < 8; ++i) acc[t][i] = 0.f;
  const float* wbase = wgt + (size_t)b * KPAD * PPAD + pidx;
  const float* tbase = all_t + (size_t)b * KPAD * CC + cbase + n;
  for (int k0 = 0; k0 < KPAD; k0 += 4) {
    const int kk = k0 + kp;
    v2f a;
    a.x = wbase[(size_t)kk * PPAD];            // A[p][k] = wgt[b][k][p]
    a.y = wbase[(size_t)(kk + 1) * PPAD];
    const float* trow = tbase + (size_t)kk * CC;
#pragma unroll
    for (int t = 0; t < 8; ++t) {
      v2f bb;
      bb.x = trow[t * 16];                     // B[k][c]
      bb.y = trow[t * 16 + CC];
      acc[t] = __builtin_amdgcn_wmma_f32_16x16x4_f32(false, a, false, bb,
                                                     (short)0, acc[t], false, false);
    }
  }
  const int mbase = (lane < 16) ? 0 : 8;
#pragma unroll
  for (int t = 0; t < 8; ++t) {
    for (int r2 = 0; r2 < 8; ++r2) {
      const int prow = mt * 16 + mbase + r2;
      if (prow < PTOK)
        super[((size_t)b * PTOK + prow) * CC + cbase + t * 16 + n] = acc[t][r2];
    }
  }
}

// ---------------------------------------------------------------------------
extern "C" void kernel_launch(void* const* d_in, const int* in_sizes, int n_in,
                              void* d_out, int out_size, void* d_ws, size_t ws_size,
                              hipStream_t stream) {
  const float* tokens = (const float*)d_in[0];
  const float* ax     = (const float*)d_in[1];
  const float* ay     = (const float*)d_in[2];
  const float* ln_g   = (const float*)d_in[3];
  const float* ln_b   = (const float*)d_in[4];
  const float* w1     = (const float*)d_in[5];
  const float* b1     = (const float*)d_in[6];
  const float* w2     = (const float*)d_in[7];
  const float* b2     = (const float*)d_in[8];
  const float* scale  = (const float*)d_in[9];

  float* super_out = (float*)d_out;                              // B*P*C
  float* mask_out  = (float*)d_out + (size_t)BB * PTOK * CC;     // B*L

  // workspace carve-up (floats)
  float* W = (float*)d_ws;
  size_t off = 0;
  float* wbuf    = W + off; off += (size_t)BB * LL;              // softmax weights
  int*   rankbuf = (int*)(W + off); off += (size_t)BB * LL;      // ranks
  float* all_t   = W + off; off += (size_t)BB * KPAD * CC;       // padded tokens
  float* mu      = W + off; off += (size_t)BB * KP1;
  float* rs      = W + off; off += (size_t)BB * KP1;
  float* w1pk    = W + off; off += (size_t)CC * HP;              // k-pair packed
  float* lnp     = W + off; off += (size_t)CC * 2;               // g/b interleave
  float* b1p     = W + off; off += 128;
  float* w2pk    = W + off; off += (size_t)HP * PPAD;            // k-pair packed
  float* b2p     = W + off; off += (size_t)PPAD;
  float* hbuf    = W + off; off += (size_t)BB * KP1 * HP + 16 * HP;  // +pad rows
  float* wgt     = W + off; off += (size_t)BB * KPAD * PPAD;

  // 1. pack/pad weights into WMMA-friendly k-pair layouts; zero hbuf pad rows
  {
    int total = CC * HP + CC * 2 + HP + HP * PPAD + PPAD + 16 * HP;
    tam_pack_kernel<<<(total + 255) / 256, 256, 0, stream>>>(
        w1, b1, w2, b2, ln_g, ln_b, w1pk, lnp, b1p, w2pk, b2p,
        hbuf + (size_t)BB * KP1 * HP);
  }
  // 2. exact stable top-K ranks + mask + non-kept softmax weights
  tam_rank_kernel<<<BB, 1024, 0, stream>>>(ax, ay, rankbuf, wbuf, mask_out);
  // 3. gather kept rows into slot=rank, fused LN stats
  tam_gather_kernel<<<BB * LL, 128, 0, stream>>>(tokens, rankbuf, all_t, mu, rs);
  // 4. extra token (weighted reduction over non-kept)
  tam_extra_kernel<<<dim3(BB, CC / 64), 256, 0, stream>>>(tokens, wbuf, all_t);
  // 5. extra-row LN stats + zero padded rows
  tam_extra_stats_kernel<<<BB, 128, 0, stream>>>(all_t, mu, rs);
  // 6. GEMM1 (+LN fused on A, +bias +gelu); one wave = 16 rows x 112 cols
  tam_gemm1_kernel<<<dim3((KPAD / 16 + 7) / 8, BB), 256, 0, stream>>>(
      all_t, mu, rs, lnp, w1pk, b1p, hbuf);
  // 7. GEMM2 (+bias, *scale); one wave = 16 rows x 64 cols
  {
    const int tiles = (KPAD / 16) * ((PPAD / 16) / 4);           // 1677
    tam_gemm2_kernel<<<dim3((tiles + 7) / 8, BB), 256, 0, stream>>>(
        hbuf, w2pk, b2p, scale, wgt);
  }
  // 8. softmax over token axis (also zeroes padded rows)
  tam_softmax_kernel<<<dim3(PPAD / 64, BB), 256, 0, stream>>>(wgt);
  // 9. final einsum super = wgt^T @ all_tokens; one wave = 16 p x 128 c
  {
    const int tiles = (PPAD / 16) * ((CC / 16) / 8);             // 208
    tam_einsum_kernel<<<dim3((tiles + 7) / 8, BB), 256, 0, stream>>>(
        wgt, all_t, super_out);
  }
  (void)in_sizes; (void)n_in; (void)out_size; (void)ws_size;
}